// GNN_39230231282214
// MI455X (gfx1250) — compile-verified
//
#include <hip/hip_runtime.h>

typedef float v2f __attribute__((ext_vector_type(2)));
typedef float v8f __attribute__((ext_vector_type(8)));

#define NNODES 50000
#define D_IN   100
#define D_HID  128

// ---------------- zero workspace (cnt + agg1 + agg2 are contiguous) --------
__global__ __launch_bounds__(256) void zero_kernel(float* __restrict__ p, long n) {
    long i = (long)blockIdx.x * blockDim.x + threadIdx.x;
    long stride = (long)gridDim.x * blockDim.x;
    for (; i < n; i += stride) p[i] = 0.0f;
}

// ---------------- degree count ---------------------------------------------
__global__ __launch_bounds__(256) void count_kernel(const int* __restrict__ dst,
                                                    float* __restrict__ cnt, int E) {
    int e = blockIdx.x * blockDim.x + threadIdx.x;
    if (e < E) atomicAdd(cnt + dst[e], 1.0f);
}

__global__ __launch_bounds__(256) void invcnt_kernel(float* __restrict__ cnt, int N) {
    int i = blockIdx.x * blockDim.x + threadIdx.x;
    if (i < N) cnt[i] = 1.0f / fmaxf(cnt[i], 1.0f);
}

// ---------------- pack W[K][128] into WMMA B-fragment order ----------------
// Wp[(k4*2 + hi)*128 + n] = { W[(4*k4 + 2*hi)][n], W[(4*k4 + 2*hi + 1)][n] }
// so a lane's B fragment for k-step k4 is ONE coalesced b64 load.
__global__ __launch_bounds__(256) void packW_kernel(const float* __restrict__ W,
                                                    float2* __restrict__ Wp, int K) {
    int idx = blockIdx.x * blockDim.x + threadIdx.x;
    int total = (K / 4) * 2 * 128;
    if (idx >= total) return;
    int n  = idx & 127;
    int hi = (idx >> 7) & 1;
    int k4 = idx >> 8;
    float2 v;
    v.x = W[(size_t)(4 * k4 + 2 * hi + 0) * D_HID + n];
    v.y = W[(size_t)(4 * k4 + 2 * hi + 1) * D_HID + n];
    Wp[idx] = v;
}

// ---------------- edge scatter: AGG[dst] += X[src], float4 granularity -----
// 32 lanes per edge; lane c handles features [4c, 4c+4); nchunk = F/4.
__global__ __launch_bounds__(256) void scatter_kernel(const int* __restrict__ src,
                                                      const int* __restrict__ dst,
                                                      const float* __restrict__ X,
                                                      float* __restrict__ AGG,
                                                      int E, int F, int nchunk) {
    long t = (long)blockIdx.x * blockDim.x + threadIdx.x;
    int e = (int)(t >> 5);
    int c = (int)(t & 31);
    if (e >= E || c >= nchunk) return;
    int s = src[e];
    int d = dst[e];
    const float4 v = *(const float4*)(X + (size_t)s * F + c * 4);
    float* p = AGG + (size_t)d * F + c * 4;
    atomicAdd(p + 0, v.x);
    atomicAdd(p + 1, v.y);
    atomicAdd(p + 2, v.z);
    atomicAdd(p + 3, v.w);
}

// ---------------- fused SAGE GEMM ------------------------------------------
// OUT[M,128] = act( (AGG .* invcnt_row) @ WL  +  X @ WR  +  bias )
// One wave computes a 16x64 strip (4 N-tiles) via V_WMMA_F32_16X16X4_F32:
// A fragment is loaded ONCE per k-step and reused by 4 WMMAs (4x less A
// traffic than tile-per-wave), B fragments come from the packed weights as
// single b64 loads. fp32-exact; mean division folded into the A fragment.
__global__ __launch_bounds__(256) void sage_gemm_kernel(
        const float* __restrict__ AGG, const float* __restrict__ invcnt,
        const float* __restrict__ X,
        const float2* __restrict__ WLp, const float2* __restrict__ WRp,
        const float* __restrict__ bias,
        float* __restrict__ OUT, int M, int K, int doRelu) {
    const int lane = threadIdx.x & 31;
    const int w    = blockIdx.x * (blockDim.x >> 5) + (threadIdx.x >> 5);
    const int half = w & 1;            // 64-column half: cols [half*64, half*64+64)
    const int tM   = w >> 1;
    if (tM * 16 >= M) return;          // wave-uniform: EXEC all-ones for WMMA
    const int r  = lane & 15;
    const int hi = lane >> 4;
    const int m  = tM * 16 + r;
    const int nb = half * 64 + r;      // lane's column within tile t: nb + t*16
    const float ic = invcnt[m];

    v8f acc[4] = {};

    const float* arow = AGG + (size_t)m * K;
    const float* xrow = X + (size_t)m * K;
    const int K4 = K >> 2;

    for (int k4 = 0; k4 < K4; ++k4) {
        v2f a = *(const v2f*)(arow + k4 * 4 + 2 * hi);
        a.x *= ic; a.y *= ic;          // fold mean division into A fragment
        const float2* bp = WLp + (size_t)(k4 * 2 + hi) * 128 + nb;
#pragma unroll
        for (int t = 0; t < 4; ++t) {
            v2f b = *(const v2f*)(bp + t * 16);
            acc[t] = __builtin_amdgcn_wmma_f32_16x16x4_f32(false, a, false, b,
                                                           (short)0, acc[t],
                                                           false, false);
        }
    }
    for (int k4 = 0; k4 < K4; ++k4) {
        v2f a = *(const v2f*)(xrow + k4 * 4 + 2 * hi);
        const float2* bp = WRp + (size_t)(k4 * 2 + hi) * 128 + nb;
#pragma unroll
        for (int t = 0; t < 4; ++t) {
            v2f b = *(const v2f*)(bp + t * 16);
            acc[t] = __builtin_amdgcn_wmma_f32_16x16x4_f32(false, a, false, b,
                                                           (short)0, acc[t],
                                                           false, false);
        }
    }

#pragma unroll
    for (int t = 0; t < 4; ++t) {
        const int n = nb + t * 16;
        const float bv = bias[n];
#pragma unroll
        for (int j = 0; j < 8; ++j) {
            float v = acc[t][j] + bv;
            if (doRelu) v = fmaxf(v, 0.0f);
            OUT[(size_t)(tM * 16 + j + 8 * hi) * D_HID + n] = v;
        }
    }
}

extern "C" void kernel_launch(void* const* d_in, const int* in_sizes, int n_in,
                              void* d_out, int out_size, void* d_ws, size_t ws_size,
                              hipStream_t stream) {
    (void)n_in; (void)out_size; (void)ws_size;
    const float* x   = (const float*)d_in[0];
    const int*   ei  = (const int*)d_in[1];   // [2, E] int32 (JAX default, no x64)
    const float* Wl1 = (const float*)d_in[2];
    const float* bl1 = (const float*)d_in[3];
    const float* Wr1 = (const float*)d_in[4];
    const float* Wl2 = (const float*)d_in[5];
    const float* bl2 = (const float*)d_in[6];
    const float* Wr2 = (const float*)d_in[7];

    const int E = in_sizes[1] / 2;
    const int N = NNODES;
    const int* src = ei;
    const int* dst = ei + E;

    float* ws   = (float*)d_ws;
    float* cnt  = ws;                                  // N        (becomes invcnt)
    float* agg1 = cnt  + N;                            // N * 100
    float* agg2 = agg1 + (size_t)N * D_IN;             // N * 128
    float* h    = agg2 + (size_t)N * D_HID;            // N * 128
    float*  wp   = h   + (size_t)N * D_HID;
    float2* Wl1p = (float2*)wp;                        // (100/4)*2*128 = 6400 float2
    float2* Wr1p = Wl1p + 6400;
    float2* Wl2p = Wr1p + 6400;                        // (128/4)*2*128 = 8192 float2
    float2* Wr2p = Wl2p + 8192;
    float* out  = (float*)d_out;

    // 1) zero cnt + agg1 + agg2 (contiguous) — must happen every call
    const long zn = (long)N + (long)N * D_IN + (long)N * D_HID;
    zero_kernel<<<2048, 256, 0, stream>>>(cnt, zn);

    // 2) pack weights into WMMA B-fragment order (tiny)
    packW_kernel<<<(6400 + 255) / 256, 256, 0, stream>>>(Wl1, Wl1p, D_IN);
    packW_kernel<<<(6400 + 255) / 256, 256, 0, stream>>>(Wr1, Wr1p, D_IN);
    packW_kernel<<<(8192 + 255) / 256, 256, 0, stream>>>(Wl2, Wl2p, D_HID);
    packW_kernel<<<(8192 + 255) / 256, 256, 0, stream>>>(Wr2, Wr2p, D_HID);

    // 3) degree -> 1/max(deg,1)
    count_kernel<<<(E + 255) / 256, 256, 0, stream>>>(dst, cnt, E);
    invcnt_kernel<<<(N + 255) / 256, 256, 0, stream>>>(cnt, N);

    // 4) layer-1 aggregation (F=100, 25 float4 chunks of 32 lane slots)
    const long st = (long)E * 32;
    const int  sgrid = (int)((st + 255) / 256);
    scatter_kernel<<<sgrid, 256, 0, stream>>>(src, dst, x, agg1, E, D_IN, 25);

    // 5) layer-1 fused GEMM + bias + ReLU -> h
    const int mtiles = N / 16;                 // 3125, exact
    const int waves  = mtiles * 2;             // 2 half-strips of 64 cols
    const int ggrid  = (waves + 7) / 8;        // 8 waves / 256-thread block
    sage_gemm_kernel<<<ggrid, 256, 0, stream>>>(agg1, cnt, x, Wl1p, Wr1p, bl1,
                                                h, N, D_IN, 1);

    // 6) layer-2 aggregation (F=128, 32 chunks)
    scatter_kernel<<<sgrid, 256, 0, stream>>>(src, dst, h, agg2, E, D_HID, 32);

    // 7) layer-2 fused GEMM + bias -> out
    sage_gemm_kernel<<<ggrid, 256, 0, stream>>>(agg2, cnt, h, Wl2p, Wr2p, bl2,
                                                out, N, D_HID, 0);
}